// CartesianEquivariantBlock_16810501997279
// MI455X (gfx1250) — compile-verified
//
#include <hip/hip_runtime.h>

typedef float v2f __attribute__((ext_vector_type(2)));
typedef float v4f __attribute__((ext_vector_type(4)));
typedef float v8f __attribute__((ext_vector_type(8)));

#define NCH        64
#define TILE_NODES 16
#define NSLICE     65
#define NWAVES     8
#define THREADS    (NWAVES * 32)

// LDS layout (in floats). Padded node strides (+4) keep 16B alignment and
// spread the 16 WMMA-A lanes across 16 distinct banks (stride % 64 == 4).
#define MIXED_SZ   (NSLICE * TILE_NODES * NCH)   // 66560 floats
#define A0_STR     68
#define A1_STR     196
#define A2_STR     580
#define A0_OFF     MIXED_SZ
#define A1_OFF     (A0_OFF + TILE_NODES * A0_STR)
#define A2_OFF     (A1_OFF + TILE_NODES * A1_STR)
#define LDS_FLOATS (A2_OFF + TILE_NODES * A2_STR) // 80064 floats = 320256 B

#define GAS __attribute__((address_space(1)))

__device__ __forceinline__ float ldg (const float* p) { return *(const GAS float*)p; }
__device__ __forceinline__ v2f   ldg2(const float* p) { return *(const GAS v2f*)p; }
__device__ __forceinline__ v4f   ldg4(const float* p) { return *(const GAS v4f*)p; }
__device__ __forceinline__ void  stg_nt(float v, float* p) {
  __builtin_nontemporal_store(v, (GAS float*)p);
}

// Per-group metadata. Group g covers slices built from source tensor a<g>.
// group0 (a0): 5 scalar slices; group1 (a1): 5 vector slice-triples;
// group2 (a2): 5 tensor slice-9-blocks.
__constant__ unsigned char g0_slice[5] = { 0, 13, 14, 15, 19 };
__constant__ unsigned char g0_w[5]     = { 0,  3,  4,  5,  7 };
__constant__ unsigned char g1_base[5]  = { 1, 16, 29, 32, 35 };
__constant__ unsigned char g1_w[5]     = { 1,  6,  9, 10, 11 };
__constant__ unsigned char g2_base[5]  = { 4, 20, 38, 47, 56 };
__constant__ unsigned char g2_w[5]     = { 2,  8, 12, 13, 14 };

// K=64 WMMA chain with compile-time A stride: all ds_load / global_load
// offsets become instruction immediates.
template<int CS>
__device__ __forceinline__ v8f mix_gemm(const float* __restrict__ asrc,
                                        const float* __restrict__ wrow) {
  v8f acc = { 0.f, 0.f, 0.f, 0.f, 0.f, 0.f, 0.f, 0.f };
#pragma unroll
  for (int kk = 0; kk < 16; ++kk) {
    v2f a, b;
    if (CS == 1) {
      a = *(const v2f*)(asrc + kk * 4);          // ds_load_b64, imm offset
    } else {
      a.x = asrc[(kk * 4    ) * CS];             // ds_load_b32, imm offset
      a.y = asrc[(kk * 4 + 1) * CS];
    }
    b = ldg2(wrow + kk * 4);                     // global_load_b64, imm offset
    acc = __builtin_amdgcn_wmma_f32_16x16x4_f32(
        /*neg_a=*/false, a, /*neg_b=*/false, b,
        /*c_mod=*/(short)0, acc, /*reuse_a=*/false, /*reuse_b=*/false);
  }
  return acc;
}

// Scatter a 16x16 f32 D tile (WMMA C/D layout) into mixed[slice] in LDS.
__device__ __forceinline__ void store_tile(float* __restrict__ smem, int slice,
                                           int cb, int mrow, int lane, v8f acc) {
  float* dst = smem + slice * (TILE_NODES * NCH) + cb * 16 + mrow;
  const int rofs = (lane >> 4) * 8;   // lanes 0-15: rows v; lanes 16-31: rows v+8
#pragma unroll
  for (int v = 0; v < 8; ++v)
    dst[(v + rofs) * NCH] = acc[v];
}

extern "C" __global__ __launch_bounds__(THREADS)
void cart_equiv_fused_kernel(
    const float* __restrict__ a0, const float* __restrict__ a1, const float* __restrict__ a2,
    const float* __restrict__ m0,  const float* __restrict__ m1,  const float* __restrict__ m2,
    const float* __restrict__ m00, const float* __restrict__ m01, const float* __restrict__ m02,
    const float* __restrict__ m11, const float* __restrict__ m12, const float* __restrict__ m22,
    const float* __restrict__ pw0, const float* __restrict__ pw1, const float* __restrict__ pw2,
    float* __restrict__ out, int N)
{
  extern __shared__ float smem[];

  const int tid   = threadIdx.x;
  const int lane  = tid & 31;
  const int wave  = tid >> 5;
  const int mrow  = lane & 15;          // WMMA row (node) / B column (out-ch)
  const int koff  = (lane >> 4) << 1;   // lane-half K offset: 0 or 2
  const int node0 = blockIdx.x * TILE_NODES;

  // ------------- Stage raw input tile into LDS (coalesced b128 loads) -------------
  {
    { // a0 tile: 1024 floats = 256 float4 -> 1 per thread
      const int f = tid * 4;
      const int n = f >> 6, r = f & 63;
      int zz = node0 + n; if (zz >= N) zz = N - 1;
      *(v4f*)(smem + A0_OFF + n * A0_STR + r) = ldg4(a0 + zz * 64 + r);
    }
#pragma unroll
    for (int j = 0; j < 3; ++j) { // a1 tile: 3072 floats
      const int f = (j * THREADS + tid) * 4;
      const int n = f / 192, r = f % 192;
      int zz = node0 + n; if (zz >= N) zz = N - 1;
      *(v4f*)(smem + A1_OFF + n * A1_STR + r) = ldg4(a1 + zz * 192 + r);
    }
#pragma unroll
    for (int j = 0; j < 9; ++j) { // a2 tile: 9216 floats
      const int f = (j * THREADS + tid) * 4;
      const int n = f / 576, r = f % 576;
      int zz = node0 + n; if (zz >= N) zz = N - 1;
      *(v4f*)(smem + A2_OFF + n * A2_STR + r) = ldg4(a2 + zz * 576 + r);
    }
  }
  __syncthreads();

  // ------------- Phase 1: channel-mixing GEMMs via f32 WMMA -------------
  const float* Wp[15] = {
    m0, m1, m2,
    m00, m00 + 4096, m01, m01 + 4096, m02, m02 + 4096,
    m11, m11 + 4096, m12, m12 + 4096, m22, m22 + 4096
  };
  const int wcol = mrow * NCH + koff;   // per-lane offset into W row block

  // group 0: source a0 (CS=1), 5 slices x 4 cb = 20 jobs
  for (int j = wave; j < 5 * 4; j += NWAVES) {
    const int si = j >> 2, cb = j & 3;
    const float* wrow = Wp[g0_w[si]] + cb * 16 * NCH + wcol;
    const float* asrc = smem + A0_OFF + mrow * A0_STR + koff * 1;
    v8f acc = mix_gemm<1>(asrc, wrow);
    store_tile(smem, g0_slice[si], cb, mrow, lane, acc);
  }
  // group 1: source a1 (CS=3), 15 slices x 4 cb = 60 jobs
  for (int j = wave; j < 15 * 4; j += NWAVES) {
    const int si = j >> 2, cb = j & 3;
    const int q = si / 3, cm = si % 3;
    const float* wrow = Wp[g1_w[q]] + cb * 16 * NCH + wcol;
    const float* asrc = smem + A1_OFF + mrow * A1_STR + cm + koff * 3;
    v8f acc = mix_gemm<3>(asrc, wrow);
    store_tile(smem, g1_base[q] + cm, cb, mrow, lane, acc);
  }
  // group 2: source a2 (CS=9), 45 slices x 4 cb = 180 jobs
  for (int j = wave; j < 45 * 4; j += NWAVES) {
    const int si = j >> 2, cb = j & 3;
    const int q = si / 9, cm = si % 9;
    const float* wrow = Wp[g2_w[q]] + cb * 16 * NCH + wcol;
    const float* asrc = smem + A2_OFF + mrow * A2_STR + cm + koff * 9;
    v8f acc = mix_gemm<9>(asrc, wrow);
    store_tile(smem, g2_base[q] + cm, cb, mrow, lane, acc);
  }

  __syncthreads();

  // ------------- Phase 2: path contractions + path-weight reduction -------------
  const int c = tid & (NCH - 1);           // rep-invariant channel
  float wp0[8], wp1[5], wp2[9];
#pragma unroll
  for (int p = 0; p < 8; ++p) wp0[p] = ldg(pw0 + p * NCH + c);
#pragma unroll
  for (int p = 0; p < 5; ++p) wp1[p] = ldg(pw1 + p * NCH + c);
#pragma unroll
  for (int p = 0; p < 9; ++p) wp2[p] = ldg(pw2 + p * NCH + c);

  const int base1 = N * NCH;      // msg1 offset in flat output
  const int base2 = N * NCH * 4;  // msg2 offset

#pragma unroll
  for (int rep = 0; rep < (TILE_NODES * NCH) / THREADS; ++rep) {
    const int zl = rep * (THREADS / NCH) + (tid >> 6);
    const int z  = node0 + zl;

    float Mv[NSLICE];
#pragma unroll
    for (int s = 0; s < NSLICE; ++s)
      Mv[s] = smem[s * (TILE_NODES * NCH) + zl * NCH + c];

    if (z >= N) continue;

    const float  s0v  = Mv[0];
    const float* v1   = &Mv[1];
    const float* T2   = &Mv[4];
    const float  s00a = Mv[13], s00b = Mv[14], s01 = Mv[15];
    const float* v01  = &Mv[16];
    const float  s02  = Mv[19];
    const float* T02  = &Mv[20];
    const float* v11a = &Mv[29];
    const float* v11b = &Mv[32];
    const float* v12  = &Mv[35];
    const float* T12  = &Mv[38];
    const float* TA   = &Mv[47];   // T22A
    const float* TB   = &Mv[56];   // T22B

    const float trA   = TA[0] + TA[4] + TA[8];
    const float trB   = TB[0] + TB[4] + TB[8];
    const float trT02 = T02[0] + T02[4] + T02[8];
    const float trT12 = T12[0] + T12[4] + T12[8];
    const float tr2   = T2[0] + T2[4] + T2[8];

    // ---- bucket 0 (8 paths, reference enumeration order) ----
    float fro = 0.f, frt = 0.f;
#pragma unroll
    for (int d = 0; d < 3; ++d)
#pragma unroll
      for (int e = 0; e < 3; ++e) {
        fro += TA[d * 3 + e] * TB[d * 3 + e];   // cons {(0,2),(1,3)}
        frt += TA[d * 3 + e] * TB[e * 3 + d];   // cons {(0,3),(1,2)}
      }
    const float dot11 = v11a[0] * v11b[0] + v11a[1] * v11b[1] + v11a[2] * v11b[2];

    float o0 = 0.f;
    o0 += wp0[0] * s0v;
    o0 += wp0[1] * tr2;
    o0 += wp0[2] * (s00a * s00b);
    o0 += wp0[3] * (s02 * trT02);
    o0 += wp0[4] * dot11;
    o0 += wp0[5] * (trA * trB);
    o0 += wp0[6] * fro;
    o0 += wp0[7] * frt;
    stg_nt(o0, out + z * NCH + c);

    // ---- bucket 1 (5 paths) ----
#pragma unroll
    for (int d = 0; d < 3; ++d) {
      float mtu = v12[0] * T12[0 * 3 + d] + v12[1] * T12[1 * 3 + d] + v12[2] * T12[2 * 3 + d]; // cons (0,1)
      float mu  = v12[0] * T12[d * 3 + 0] + v12[1] * T12[d * 3 + 1] + v12[2] * T12[d * 3 + 2]; // cons (0,2)
      float o1  = wp1[0] * v1[d]
                + wp1[1] * (s01 * v01[d])
                + wp1[2] * mtu
                + wp1[3] * mu
                + wp1[4] * (v12[d] * trT12);                                                   // cons (1,2)
      stg_nt(o1, out + base1 + (z * NCH + c) * 3 + d);
    }

    // ---- bucket 2 (9 paths) ----
#pragma unroll
    for (int d = 0; d < 3; ++d)
#pragma unroll
      for (int e = 0; e < 3; ++e) {
        float r4 = 0.f, r5 = 0.f, r6 = 0.f, r7 = 0.f;
#pragma unroll
        for (int k = 0; k < 3; ++k) {
          r4 += TA[k * 3 + d] * TB[k * 3 + e];  // cons (0,2): A^T B
          r5 += TA[k * 3 + d] * TB[e * 3 + k];  // cons (0,3): A^T B^T
          r6 += TA[d * 3 + k] * TB[k * 3 + e];  // cons (1,2): A B
          r7 += TA[d * 3 + k] * TB[e * 3 + k];  // cons (1,3): A B^T
        }
        float o2 = wp2[0] * T2[d * 3 + e]
                 + wp2[1] * (s02 * T02[d * 3 + e])
                 + wp2[2] * (v11a[d] * v11b[e])
                 + wp2[3] * (trA * TB[d * 3 + e])   // cons (0,1)
                 + wp2[4] * r4
                 + wp2[5] * r5
                 + wp2[6] * r6
                 + wp2[7] * r7
                 + wp2[8] * (trB * TA[d * 3 + e]);  // cons (2,3)
        stg_nt(o2, out + base2 + (z * NCH + c) * 9 + d * 3 + e);
      }
  }
}

extern "C" void kernel_launch(void* const* d_in, const int* in_sizes, int n_in,
                              void* d_out, int out_size, void* d_ws, size_t ws_size,
                              hipStream_t stream) {
  (void)n_in; (void)out_size; (void)d_ws; (void)ws_size;
  const float* a0  = (const float*)d_in[0];
  const float* a1  = (const float*)d_in[1];
  const float* a2  = (const float*)d_in[2];
  const float* m0  = (const float*)d_in[3];
  const float* m1  = (const float*)d_in[4];
  const float* m2  = (const float*)d_in[5];
  const float* m00 = (const float*)d_in[6];
  const float* m01 = (const float*)d_in[7];
  const float* m02 = (const float*)d_in[8];
  const float* m11 = (const float*)d_in[9];
  const float* m12 = (const float*)d_in[10];
  const float* m22 = (const float*)d_in[11];
  const float* pw0 = (const float*)d_in[12];
  const float* pw1 = (const float*)d_in[13];
  const float* pw2 = (const float*)d_in[14];
  float* out = (float*)d_out;

  const int N = in_sizes[0] / NCH;                       // 30000
  const int blocks = (N + TILE_NODES - 1) / TILE_NODES;  // 1875
  const size_t shmem = (size_t)LDS_FLOATS * sizeof(float); // 320256 B

  cart_equiv_fused_kernel<<<blocks, THREADS, shmem, stream>>>(
      a0, a1, a2, m0, m1, m2, m00, m01, m02, m11, m12, m22,
      pw0, pw1, pw2, out, N);
}